// SparseLinear_40261023433168
// MI455X (gfx1250) — compile-verified
//
#include <hip/hip_runtime.h>

typedef __attribute__((ext_vector_type(2))) float v2f;
typedef __attribute__((ext_vector_type(4))) float v4f;
typedef __attribute__((ext_vector_type(8))) float v8f;
typedef __attribute__((ext_vector_type(4))) int   v4i;

#define M_DIM 8192   // IN_FEATURES  (rows of W, rows of out)
#define K_DIM 4096   // OUT_FEATURES (cols of W, rows of x)
#define N_DIM 4096   // N_COLS
#define NNZ_N 4194304

#define BM 128
#define BN 128
#define KT 16
#define NKT (K_DIM / KT)
#define LDA 20    // A-tile LDS stride (floats): 16B-aligned rows, conflict-free b64 reads
#define PSB 288   // B pair-row LDS stride (floats): 2*BN + 32 -> lane halves on disjoint banks

// Use the CDNA5 async memory->LDS path when the toolchain exposes it.
#if defined(__gfx1250__) &&                                          \
    __has_builtin(__builtin_amdgcn_global_load_async_to_lds_b128) && \
    __has_builtin(__builtin_amdgcn_s_wait_asynccnt)
#define USE_ASYNC_LDS 1
#else
#define USE_ASYNC_LDS 0
#endif

#if USE_ASYNC_LDS
__device__ __forceinline__ void async_copy16(const float* g, float* l) {
  float* gm = const_cast<float*>(g);
  __builtin_amdgcn_global_load_async_to_lds_b128(
      (__attribute__((address_space(1))) v4i*)gm,
      (__attribute__((address_space(3))) v4i*)l,
      /*offset=*/0, /*cpol=*/0);
}
#endif

// ---------------------------------------------------------------------------
// 1) Zero the dense W workspace (8192 x 4096 f32) with float4 stores.
// ---------------------------------------------------------------------------
__global__ __launch_bounds__(256) void zero_w_kernel(float* __restrict__ W) {
  size_t i = (size_t)blockIdx.x * 256 + threadIdx.x;
  v4f z = {0.f, 0.f, 0.f, 0.f};
  ((v4f*)W)[i] = z;
}

// ---------------------------------------------------------------------------
// 2) COO scatter: W[rows[i]][cols[i]] += values[i]  (duplicates sum -> atomic)
// ---------------------------------------------------------------------------
__global__ __launch_bounds__(256) void scatter_kernel(const float* __restrict__ values,
                                                      const int*   __restrict__ rows,
                                                      const int*   __restrict__ cols,
                                                      float*       __restrict__ W) {
  int i = blockIdx.x * 256 + threadIdx.x;
  int r = rows[i];
  int c = cols[i];
  size_t idx = (size_t)r * K_DIM + (size_t)c;
  __hip_atomic_fetch_add(&W[idx], values[i], __ATOMIC_RELAXED, __HIP_MEMORY_SCOPE_AGENT);
}

// ---------------------------------------------------------------------------
// 3) out = bias + W @ x with V_WMMA_F32_16X16X4_F32.
//    256 threads (8 waves) per 128x128 block, K-tile 16, double-buffered LDS.
//    A tile: async-to-LDS (row-major, padded).  B tile: K-paired LDS layout
//    ((k,n) -> [(k>>1)*PSB + 2n + (k&1)]) so every WMMA B fragment is one
//    aligned ds_load_b64 (no VGPR re-pairing movs); B staged via dual-row
//    register interleave, software-pipelined across the compute phase.
// ---------------------------------------------------------------------------
__global__ __launch_bounds__(256) void spmm_wmma_kernel(const float* __restrict__ W,
                                                        const float* __restrict__ X,
                                                        const float* __restrict__ bias,
                                                        float*       __restrict__ out) {
  __shared__ float As[2][BM * LDA];   // 2 x (128 x 16) tile of W
  __shared__ float Bs[2][8 * PSB];    // 2 x (16 x 128) tile of x, K-paired

  const int tid  = threadIdx.x;
  const int wave = tid >> 5;
  const int lane = tid & 31;
  const int l16  = lane & 15;
  const int hi   = lane >> 4;         // 0: lanes 0-15, 1: lanes 16-31

  const int nBlock = blockIdx.x * BN;
  const int mBlock = blockIdx.y * BM;

  // ---- A staging coords: 2 x 16B per thread per tile (async path)
  const int ar = tid >> 2;            // rows 0..63 (+64 for second chunk)
  const int ac = (tid & 3) << 2;      // col group within 16-wide tile
  const float* gA0 = W + (size_t)(mBlock + ar) * K_DIM + ac;
  const float* gA1 = W + (size_t)(mBlock + ar + 64) * K_DIM + ac;
  float* lA0base = &As[0][ar * LDA + ac];
  float* lA1base = &As[0][(ar + 64) * LDA + ac];
  const int ldsStrideA = BM * LDA;

  // ---- B staging coords: one (pair-row, 4-col) cell per thread
  const int k2 = tid >> 5;            // pair-row 0..7 (k = 2*k2, 2*k2+1)
  const int n4 = (tid & 31) << 2;     // col 0..124
  const float* gBp0 = X + (size_t)(2 * k2)     * N_DIM + (nBlock + n4);
  const float* gBp1 = X + (size_t)(2 * k2 + 1) * N_DIM + (nBlock + n4);
  float* lBbase = &Bs[0][k2 * PSB + n4 * 2];
  const int ldsStrideB = 8 * PSB;

  v8f acc[8];
  const v8f zero = {0.f, 0.f, 0.f, 0.f, 0.f, 0.f, 0.f, 0.f};
#pragma unroll
  for (int t = 0; t < 8; ++t) acc[t] = zero;

  const int aRow = wave * 16 + l16;   // A-fragment row (16x4 layout)

  // ---------------- prologue: stage tile 0 into buffer 0 ----------------
  {
    v4f r0 = *(const v4f*)gBp0;
    v4f r1 = *(const v4f*)gBp1;
#if USE_ASYNC_LDS
    async_copy16(gA0, lA0base);
    async_copy16(gA1, lA1base);
#else
    v4f ra0 = *(const v4f*)gA0;
    v4f ra1 = *(const v4f*)gA1;
    *(v4f*)lA0base = ra0;
    *(v4f*)lA1base = ra1;
#endif
    v4f z0 = {r0.x, r1.x, r0.y, r1.y};
    v4f z1 = {r0.z, r1.z, r0.w, r1.w};
    *(v4f*)(lBbase)     = z0;
    *(v4f*)(lBbase + 4) = z1;
#if USE_ASYNC_LDS
    __builtin_amdgcn_s_wait_asynccnt(0);
#endif
  }
  __syncthreads();

  // ---------------- main pipeline ----------------
  for (int kt = 0; kt < NKT; ++kt) {
    const int cur = kt & 1;
    const int nxt = cur ^ 1;
    const bool have = (kt + 1 < NKT);
    const int k0n = (kt + 1) * KT;

    // Issue next tile's transfers; they overlap the compute phase below.
    v4f r0, r1;
#if !USE_ASYNC_LDS
    v4f ra0, ra1;
#endif
    if (have) {
#if USE_ASYNC_LDS
      async_copy16(gA0 + k0n, lA0base + nxt * ldsStrideA);
      async_copy16(gA1 + k0n, lA1base + nxt * ldsStrideA);
#else
      ra0 = *(const v4f*)(gA0 + k0n);
      ra1 = *(const v4f*)(gA1 + k0n);
#endif
      r0 = *(const v4f*)(gBp0 + (size_t)k0n * N_DIM);
      r1 = *(const v4f*)(gBp1 + (size_t)k0n * N_DIM);
    }

    // ---- 4 K-substeps x 8 N-tiles: 32 WMMAs per wave on buffer `cur`
    const float* as = &As[cur][0];
    const float* bs = &Bs[cur][0];
#pragma unroll
    for (int kk = 0; kk < KT; kk += 4) {
      // A 16x4 fragment: lane holds (row = aRow, K = kk + 2*hi, +1) as float2
      v2f a = *(const v2f*)(&as[aRow * LDA + kk + 2 * hi]);
      // B pair-row for this lane: k = kk + 2*hi  ->  pair index (kk>>1) + hi
      const float* brow = &bs[((kk >> 1) + hi) * PSB + l16 * 2];
#pragma unroll
      for (int t = 0; t < 8; ++t) {
        v2f b = *(const v2f*)(&brow[t * 32]);  // (K, K+1) contiguous: one b64
        acc[t] = __builtin_amdgcn_wmma_f32_16x16x4_f32(
            /*neg_a=*/false, a, /*neg_b=*/false, b,
            /*c_mod=*/(short)0, acc[t], /*reuse_a=*/false, /*reuse_b=*/false);
      }
    }

    // Tile boundary: next tile's data must be resident before anyone reads it.
    if (have) {
#if !USE_ASYNC_LDS
      *(v4f*)(lA0base + nxt * ldsStrideA) = ra0;
      *(v4f*)(lA1base + nxt * ldsStrideA) = ra1;
#endif
      v4f z0 = {r0.x, r1.x, r0.y, r1.y};
      v4f z1 = {r0.z, r1.z, r0.w, r1.w};
      *(v4f*)(lBbase + nxt * ldsStrideB)     = z0;
      *(v4f*)(lBbase + nxt * ldsStrideB + 4) = z1;
#if USE_ASYNC_LDS
      __builtin_amdgcn_s_wait_asynccnt(0);
#endif
    }
    __syncthreads();
  }

  // ---- epilogue: C layout => VGPR r holds M = r + 8*hi, N = l16 per 16-col tile
#pragma unroll
  for (int t = 0; t < 8; ++t) {
    int nCol = nBlock + t * 16 + l16;
    float bv = bias[nCol];
#pragma unroll
    for (int r = 0; r < 8; ++r) {
      int mRow = mBlock + wave * 16 + hi * 8 + r;
      out[(size_t)mRow * N_DIM + nCol] = acc[t][r] + bv;
    }
  }
}

// ---------------------------------------------------------------------------
// Launch: zero W -> scatter COO -> WMMA GEMM (stream-ordered).
// ---------------------------------------------------------------------------
extern "C" void kernel_launch(void* const* d_in, const int* in_sizes, int n_in,
                              void* d_out, int out_size, void* d_ws, size_t ws_size,
                              hipStream_t stream) {
  (void)in_sizes; (void)n_in; (void)out_size; (void)ws_size;

  const float* values = (const float*)d_in[0];
  const float* bias   = (const float*)d_in[1];
  const float* x      = (const float*)d_in[2];
  const int*   rows   = (const int*)d_in[3];
  const int*   cols   = (const int*)d_in[4];
  float* out = (float*)d_out;
  float* W   = (float*)d_ws;  // 8192*4096*4 = 128 MB dense weight

  const int nvec4 = (M_DIM * K_DIM) / 4;               // 8,388,608 float4s
  zero_w_kernel<<<nvec4 / 256, 256, 0, stream>>>(W);

  scatter_kernel<<<NNZ_N / 256, 256, 0, stream>>>(values, rows, cols, W);

  dim3 grid(N_DIM / BN, M_DIM / BM);                   // 32 x 64 blocks
  spmm_wmma_kernel<<<grid, 256, 0, stream>>>(W, x, bias, out);
}